// RPNPostProcessor_12163347382879
// MI455X (gfx1250) — compile-verified
//
#include <hip/hip_runtime.h>
#include <cstdint>

typedef unsigned int uint;

#define NIMG 16
#define NA   3
#define NH   336
#define NW   336
#define HW   (NH * NW)        // 112896
#define HWA  (NA * HW)        // 338688 anchors / image
#define TOPK 2000
#define KPAD 2048
#define POST 1000
#define WORDS 64              // KPAD / 32
#define SLICES 32             // device-wide slices per image for scan passes
#define NMS_T 0.7f
#define BBOX_CLIP 4.135166556742356f   // log(1000/16)
#define IMMAX 1343.0f                  // IM_W - 1

// ---------------------------------------------------------------------------
// CDNA5 async global->LDS helpers (ASYNCcnt path, see CDNA5 ISA)
// ---------------------------------------------------------------------------
__device__ __forceinline__ uint32_t lds_off_of(const void* p) {
  // flat LDS address low 32 bits == offset within the workgroup LDS allocation
  return (uint32_t)(uintptr_t)p;
}
__device__ __forceinline__ void async_ld_b128(uint32_t lds_off, uint32_t voff,
                                              const void* sbase) {
  asm volatile("global_load_async_to_lds_b128 %0, %1, %2 offset:0"
               :: "v"(lds_off), "v"(voff), "s"(sbase) : "memory");
}
__device__ __forceinline__ void wait_async0() {
  asm volatile("s_wait_asynccnt 0" ::: "memory");
}

// ---------------------------------------------------------------------------
// K1: sigmoid + (N,A,H,W) -> (N, H*W*A) permute.  HBM-bound (43 MB traffic);
// use v_rcp_f32 instead of the IEEE div expansion.
// ---------------------------------------------------------------------------
__global__ void rpn_sigmoid_permute(const float* __restrict__ obj,
                                    float* __restrict__ scores) {
  int gid = blockIdx.x * blockDim.x + threadIdx.x;
  const int total = NIMG * NA * HW;
  if (gid >= total) return;
  int n  = gid / (NA * HW);
  int r  = gid - n * (NA * HW);
  int a  = r / HW;
  int hw = r - a * HW;
  float x = obj[gid];
  float s = __builtin_amdgcn_rcpf(1.0f + __expf(-x));
  scores[(size_t)n * HWA + (size_t)hw * NA + a] = s;
}

// ---------------------------------------------------------------------------
// Radix-select state: gsel[n] = {prefix, remain, takenAll, takenEq}
// ---------------------------------------------------------------------------
__global__ void rpn_init(uint* __restrict__ ghist, uint* __restrict__ gsel) {
  int g = blockIdx.x * blockDim.x + threadIdx.x;
  if (g < NIMG * 256) ghist[g] = 0u;
  if (g < NIMG) {
    gsel[g * 4 + 0] = 0u;    // prefix
    gsel[g * 4 + 1] = TOPK;  // remain
    gsel[g * 4 + 2] = 0u;    // takenAll
    gsel[g * 4 + 3] = 0u;    // takenEq
  }
}

// Device-wide histogram pass: grid (SLICES, NIMG) x 256.  Scores are sigmoid
// outputs in (0,1) => positive => IEEE bit order == value order.
__global__ void rpn_hist_pass(const float* __restrict__ scores,
                              const uint* __restrict__ gsel,
                              uint* __restrict__ ghist, int pass) {
  const int n = blockIdx.y, slice = blockIdx.x, tid = threadIdx.x;
  __shared__ uint h[256];
  h[tid] = 0u;
  __syncthreads();
  const uint pre    = gsel[n * 4 + 0];
  const int  shift  = pass * 8;
  const uint maskhi = (pass == 3) ? 0u : (0xFFFFFFFFu << ((pass + 1) * 8));
  const float* sc = scores + (size_t)n * HWA;
  const int per = (HWA + SLICES - 1) / SLICES;
  const int i0 = slice * per;
  int i1 = i0 + per; if (i1 > HWA) i1 = HWA;
  for (int i = i0 + tid; i < i1; i += 256) {
    uint k = __float_as_uint(sc[i]);
    if ((k & maskhi) == (pre & maskhi))
      atomicAdd(&h[(k >> shift) & 255u], 1u);
  }
  __syncthreads();
  if (h[tid]) atomicAdd(&ghist[n * 256 + tid], h[tid]);
}

// Pick the digit for this pass, reset histogram for the next pass.
__global__ void rpn_choose(uint* __restrict__ gsel, uint* __restrict__ ghist,
                           int pass) {
  const int n = blockIdx.x, tid = threadIdx.x;
  __shared__ uint h[256];
  h[tid] = ghist[n * 256 + tid];
  __syncthreads();
  if (tid == 0) {
    int remain = (int)gsel[n * 4 + 1];
    int cum = 0, b = 255;
    for (; b > 0; --b) {
      if (cum + (int)h[b] >= remain) break;
      cum += (int)h[b];
    }
    gsel[n * 4 + 0] |= ((uint)b) << (pass * 8);
    gsel[n * 4 + 1]  = (uint)(remain - cum);
  }
  ghist[n * 256 + tid] = 0u;
}

// Device-wide compaction: exactly TOPK entries per image via global cursors.
__global__ void rpn_compact(const float* __restrict__ scores,
                            uint* __restrict__ gsel,
                            float* __restrict__ sel_score,
                            int*   __restrict__ sel_idx) {
  const int n = blockIdx.y, slice = blockIdx.x, tid = threadIdx.x;
  const uint thr   = gsel[n * 4 + 0];
  const int  needEq = (int)gsel[n * 4 + 1];
  const float* sc = scores + (size_t)n * HWA;
  float* os = sel_score + (size_t)n * KPAD;
  int*   oi = sel_idx   + (size_t)n * KPAD;
  const int per = (HWA + SLICES - 1) / SLICES;
  const int i0 = slice * per;
  int i1 = i0 + per; if (i1 > HWA) i1 = HWA;
  for (int i = i0 + tid; i < i1; i += 256) {
    float v = sc[i];
    uint  k = __float_as_uint(v);
    if (k > thr) {
      uint p = atomicAdd(&gsel[n * 4 + 2], 1u);
      os[p] = v; oi[p] = i;
    } else if (k == thr) {
      uint e = atomicAdd(&gsel[n * 4 + 3], 1u);
      if ((int)e < needEq) {
        uint p = atomicAdd(&gsel[n * 4 + 2], 1u);
        os[p] = v; oi[p] = i;
      }
    }
  }
}

// ---------------------------------------------------------------------------
// K3: per-image in-LDS bitonic sort of 2048 (score, idx) pairs, descending,
// ascending-index tiebreak (matches lax.top_k ordering).  Slots >= TOPK are
// padded with -inf on load (compaction writes exactly TOPK entries).
// ---------------------------------------------------------------------------
__global__ void rpn_sort(float* __restrict__ sel_score, int* __restrict__ sel_idx) {
  const int n   = blockIdx.x;
  const int tid = threadIdx.x;
  __shared__ float ks[KPAD];
  __shared__ int   vs[KPAD];
  float* gs = sel_score + (size_t)n * KPAD;
  int*   gi = sel_idx   + (size_t)n * KPAD;

  for (int e = tid; e < KPAD; e += 1024) {
    bool live = (e < TOPK);
    ks[e] = live ? gs[e] : -3.4e38f;
    vs[e] = live ? gi[e] : 0;
  }

  for (unsigned k = 2; k <= KPAD; k <<= 1) {
    for (unsigned j = k >> 1; j > 0; j >>= 1) {
      __syncthreads();
      unsigned t = tid;
      unsigned i = ((t & ~(j - 1u)) << 1) | (t & (j - 1u));
      unsigned p = i | j;
      float ka = ks[i], kb = ks[p];
      int   va = vs[i], vb = vs[p];
      bool aBetter = (ka > kb) || (ka == kb && va < vb);   // earlier in desc order
      bool dirDesc = ((i & k) == 0u);
      bool doSwap  = dirDesc ? !aBetter : aBetter;
      if (doSwap) { ks[i] = kb; ks[p] = ka; vs[i] = vb; vs[p] = va; }
    }
  }
  __syncthreads();
  gs[tid] = ks[tid];               gi[tid] = vs[tid];
  gs[tid + 1024] = ks[tid + 1024]; gi[tid + 1024] = vs[tid + 1024];
}

// ---------------------------------------------------------------------------
// K4: gather anchors/regression for top-2000, decode + clip -> boxes[n][2048][4]
// ---------------------------------------------------------------------------
__global__ void rpn_decode(const float* __restrict__ boxreg,
                           const float* __restrict__ anchors,
                           const int*   __restrict__ sel_idx,
                           float* __restrict__ boxes) {
  int gid = blockIdx.x * blockDim.x + threadIdx.x;
  if (gid >= NIMG * KPAD) return;
  int n = gid >> 11;
  int j = gid & (KPAD - 1);
  float* ob = boxes + ((size_t)n * KPAD + j) * 4;
  if (j >= TOPK) { ob[0] = 0.f; ob[1] = 0.f; ob[2] = 0.f; ob[3] = 0.f; return; }

  int idx = sel_idx[(size_t)n * KPAD + j];
  const float4 av = *(const float4*)(anchors + ((size_t)n * HWA + idx) * 4);
  float a0 = av.x, a1 = av.y, a2 = av.z, a3 = av.w;

  int a  = idx % NA;
  int hw = idx / NA;
  const float* rg = boxreg + (size_t)n * (NA * 4 * HW);
  float dx = rg[(a * 4 + 0) * HW + hw];
  float dy = rg[(a * 4 + 1) * HW + hw];
  float dw = rg[(a * 4 + 2) * HW + hw];
  float dh = rg[(a * 4 + 3) * HW + hw];
  dw = fminf(dw, BBOX_CLIP);
  dh = fminf(dh, BBOX_CLIP);

  float w  = a2 - a0 + 1.0f;
  float h  = a3 - a1 + 1.0f;
  float cx = a0 + 0.5f * w;
  float cy = a1 + 0.5f * h;
  float pcx = dx * w + cx;
  float pcy = dy * h + cy;
  float pw  = __expf(dw) * w;
  float ph  = __expf(dh) * h;

  float x1 = pcx - 0.5f * pw;
  float y1 = pcy - 0.5f * ph;
  float x2 = pcx + 0.5f * pw - 1.0f;
  float y2 = pcy + 0.5f * ph - 1.0f;
  x1 = fminf(fmaxf(x1, 0.0f), IMMAX);
  y1 = fminf(fmaxf(y1, 0.0f), IMMAX);
  x2 = fminf(fmaxf(x2, 0.0f), IMMAX);
  y2 = fminf(fmaxf(y2, 0.0f), IMMAX);
  ob[0] = x1; ob[1] = y1; ob[2] = x2; ob[3] = y2;
}

// ---------------------------------------------------------------------------
// K5: per-image IoU suppression-mask matrix: mask[i][w] bit b set iff
// IoU(i, j=w*32+b) > 0.7 and j > i.  Boxes staged to LDS via ASYNC b128 loads.
// Division-free compare: iou>T <=> inter*(1+T) > T*(ai+aj)  (union > 0 for
// valid boxes; sign-flip cases involve only invalid boxes, already removed).
// ---------------------------------------------------------------------------
__global__ void rpn_nms_mask(const float* __restrict__ boxes,
                             uint* __restrict__ mask,
                             uint* __restrict__ valid_mask) {
  const int n   = blockIdx.x;
  const int tid = threadIdx.x;
  __shared__ __align__(16) float sbox[KPAD * 4];   // 32 KB
  __shared__ float sarea[KPAD];                     // 8 KB
  __shared__ uint  svalid[WORDS];

  const float* gb = boxes + (size_t)n * KPAD * 4;
  {
    uint r0 = tid, r1 = tid + 1024;
    async_ld_b128(lds_off_of(&sbox[r0 * 4]), r0 * 16u, (const void*)gb);
    async_ld_b128(lds_off_of(&sbox[r1 * 4]), r1 * 16u, (const void*)gb);
    wait_async0();
  }
  if (tid < WORDS) svalid[tid] = 0u;
  __syncthreads();

  for (int r = tid; r < KPAD; r += 1024) {
    float x1 = sbox[r * 4 + 0], y1 = sbox[r * 4 + 1];
    float x2 = sbox[r * 4 + 2], y2 = sbox[r * 4 + 3];
    float ww = x2 - x1 + 1.0f, hh = y2 - y1 + 1.0f;
    sarea[r] = ww * hh;
    if (r < TOPK && ww >= 0.0f && hh >= 0.0f)
      atomicOr(&svalid[r >> 5], 1u << (r & 31));
  }
  __syncthreads();
  if (tid < WORDS) valid_mask[(size_t)n * WORDS + tid] = svalid[tid];

  uint* gm = mask + (size_t)n * KPAD * WORDS;
  for (int w = tid; w < TOPK * WORDS; w += 1024) {
    int i  = w >> 6;
    int wj = w & (WORDS - 1);
    float x1i = sbox[i * 4 + 0], y1i = sbox[i * 4 + 1];
    float x2i = sbox[i * 4 + 2], y2i = sbox[i * 4 + 3];
    float ai  = sarea[i];
    uint m = 0u;
    int jbase = wj << 5;
    #pragma unroll 8
    for (int b = 0; b < 32; ++b) {
      int j = jbase + b;
      if (j > i && j < TOPK) {
        float xx1 = fmaxf(x1i, sbox[j * 4 + 0]);
        float yy1 = fmaxf(y1i, sbox[j * 4 + 1]);
        float xx2 = fminf(x2i, sbox[j * 4 + 2]);
        float yy2 = fminf(y2i, sbox[j * 4 + 3]);
        float inter = fmaxf(xx2 - xx1 + 1.0f, 0.0f) * fmaxf(yy2 - yy1 + 1.0f, 0.0f);
        if (inter * (1.0f + NMS_T) > NMS_T * (ai + sarea[j])) m |= (1u << b);
      }
    }
    gm[(size_t)i * WORDS + wj] = m;
  }
}

// ---------------------------------------------------------------------------
// K6: serial NMS scan (one wave32 over LDS bitmaps; mask rows prefetched per
// 64-row chunk via ASYNC b128 loads) + keep-rank prefix + final (N,1000,5) out.
// ---------------------------------------------------------------------------
__global__ void rpn_finalize(const uint*  __restrict__ mask,
                             const uint*  __restrict__ valid_mask,
                             const float* __restrict__ boxes,
                             const float* __restrict__ sel_score,
                             float* __restrict__ out) {
  const int n   = blockIdx.x;
  const int tid = threadIdx.x;
  __shared__ uint removed_s[WORDS];
  __shared__ uint keepw[WORDS];
  __shared__ __align__(16) uint chunk[64 * WORDS];   // 16 KB mask tile
  __shared__ uint wpre[WORDS + 1];

  if (tid < WORDS) {
    removed_s[tid] = ~valid_mask[(size_t)n * WORDS + tid]; // invalid => removed
    keepw[tid] = 0u;
  }
  __syncthreads();

  const uint* gm = mask + (size_t)n * KPAD * WORDS;
  for (int c0 = 0; c0 < TOPK; c0 += 64) {
    async_ld_b128(lds_off_of(&chunk[tid * 4]), tid * 16u,
                  (const void*)(gm + (size_t)c0 * WORDS));
    wait_async0();
    __syncthreads();

    if (tid < 32) {
      volatile uint* rm = removed_s;
      int iend = TOPK - c0; if (iend > 64) iend = 64;
      for (int ii = 0; ii < iend; ++ii) {
        int i = c0 + ii;
        uint r = rm[i >> 5];
        if (!((r >> (i & 31)) & 1u)) {
          rm[tid]      |= chunk[ii * WORDS + tid];
          rm[tid + 32] |= chunk[ii * WORDS + tid + 32];
          if (tid == 0) keepw[i >> 5] |= (1u << (i & 31));
        }
      }
    }
    __syncthreads();
  }

  if (tid == 0) {
    uint c = 0;
    for (int w = 0; w < WORDS; ++w) { wpre[w] = c; c += (uint)__popc(keepw[w]); }
    wpre[WORDS] = c;
  }
  __syncthreads();

  const int nk = (int)wpre[WORDS];
  for (int i = tid; i < TOPK; i += 1024) {
    uint w = i >> 5, b = i & 31;
    uint kw = keepw[w];
    bool k  = (kw >> b) & 1u;
    int rank = (int)wpre[w] + __popc(kw & ((1u << b) - 1u));
    int pos = k ? rank : (nk + (i - rank));   // ties at -1 => ascending index
    if (pos < POST) {
      const float* bx = boxes + ((size_t)n * KPAD + i) * 4;
      float sc = k ? sel_score[(size_t)n * KPAD + i] : 0.0f;
      float* o = out + ((size_t)n * POST + pos) * 5;
      o[0] = bx[0]; o[1] = bx[1]; o[2] = bx[2]; o[3] = bx[3]; o[4] = sc;
    }
  }
}

// ---------------------------------------------------------------------------
extern "C" void kernel_launch(void* const* d_in, const int* in_sizes, int n_in,
                              void* d_out, int out_size, void* d_ws, size_t ws_size,
                              hipStream_t stream) {
  const float* obj    = (const float*)d_in[0];  // (N, A, H, W)
  const float* boxreg = (const float*)d_in[1];  // (N, A*4, H, W)
  const float* anch   = (const float*)d_in[2];  // (N, HWA, 4)
  float* out = (float*)d_out;                   // (N, 1000, 5)

  // workspace layout (~30.8 MB)
  char* w = (char*)d_ws;
  float* scores    = (float*)w; w += (size_t)NIMG * HWA * sizeof(float);
  float* sel_score = (float*)w; w += (size_t)NIMG * KPAD * sizeof(float);
  int*   sel_idx   = (int*)w;   w += (size_t)NIMG * KPAD * sizeof(int);
  float* boxes     = (float*)w; w += (size_t)NIMG * KPAD * 4 * sizeof(float);
  uint*  valid_m   = (uint*)w;  w += (size_t)NIMG * WORDS * sizeof(uint);
  uint*  ghist     = (uint*)w;  w += (size_t)NIMG * 256 * sizeof(uint);
  uint*  gsel      = (uint*)w;  w += (size_t)NIMG * 4 * sizeof(uint);
  uint*  mask      = (uint*)w;  w += (size_t)NIMG * KPAD * WORDS * sizeof(uint);
  (void)in_sizes; (void)n_in; (void)out_size; (void)ws_size;

  const int total = NIMG * NA * HW;
  rpn_sigmoid_permute<<<(total + 255) / 256, 256, 0, stream>>>(obj, scores);
  rpn_init<<<(NIMG * 256 + 255) / 256, 256, 0, stream>>>(ghist, gsel);
  for (int p = 3; p >= 0; --p) {
    rpn_hist_pass<<<dim3(SLICES, NIMG), 256, 0, stream>>>(scores, gsel, ghist, p);
    rpn_choose<<<NIMG, 256, 0, stream>>>(gsel, ghist, p);
  }
  rpn_compact<<<dim3(SLICES, NIMG), 256, 0, stream>>>(scores, gsel, sel_score, sel_idx);
  rpn_sort<<<NIMG, 1024, 0, stream>>>(sel_score, sel_idx);
  rpn_decode<<<(NIMG * KPAD + 255) / 256, 256, 0, stream>>>(boxreg, anch, sel_idx, boxes);
  rpn_nms_mask<<<NIMG, 1024, 0, stream>>>(boxes, mask, valid_m);
  rpn_finalize<<<NIMG, 1024, 0, stream>>>(mask, valid_m, boxes, sel_score, out);
}